// MLPConcatHead_87393994539113
// MI455X (gfx1250) — compile-verified
//
#include <hip/hip_runtime.h>
#include <hip/hip_bf16.h>
#include <math.h>

typedef __attribute__((ext_vector_type(2))) float v2f;
typedef __attribute__((ext_vector_type(8))) float v8f;

#define NROW 768
#define DIM  256
#define PDIM 32
#define LDP  260   // padded LDS row stride (floats) to avoid 64-bank conflicts

__device__ __forceinline__ float mish_f(float t) {
    // mish(t) = t * tanh(softplus(t)) ; tanh(log(1+e^t)) = (u^2-1)/(u^2+1), u = 1+e^t
    float e  = __expf(fminf(t, 20.0f));   // clamp avoids inf/inf NaN; tanh==1 there anyway
    float u  = 1.0f + e;
    float u2 = u * u;
    return t * (u2 - 1.0f) * __builtin_amdgcn_rcpf(u2 + 1.0f);
}

// ---------------------------------------------------------------------------
// Kernel 1: X = relu(LN(h @ W1 + b1))   [768,256] = [768,256]@[256,256]
// One block per 16-row slab; 8 waves x 2 col-tiles, WMMA f32 16x16x4, K=256.
// ---------------------------------------------------------------------------
__global__ void __launch_bounds__(256)
k_preproj(const float* __restrict__ h, const float* __restrict__ W1,
          const float* __restrict__ b1, const float* __restrict__ g1,
          const float* __restrict__ be1, float* __restrict__ X)
{
    __shared__ float Ha[16][LDP];
    __shared__ float Xs[16][LDP];
    __shared__ float ps[16][16], pq[16][16];
    __shared__ float mu_s[16], sc_s[16];

    const int t  = threadIdx.x;
    const int i0 = blockIdx.x * 16;

    // cooperative load of 16x256 slab of h into LDS (float4, coalesced)
    for (int c = 0; c < 4; ++c) {
        int chunk = t + c * 256;          // 1024 float4 chunks total
        int row   = chunk >> 6;
        int col4  = (chunk & 63) << 2;
        *(float4*)(&Ha[row][col4]) = *(const float4*)(h + (i0 + row) * DIM + col4);
    }
    __syncthreads();

    const int l     = t & 31;
    const int w     = t >> 5;
    const int arow  = l & 15;          // A: M = lane&15
    const int kb    = (l >> 4) << 1;   // K base: lanes 16-31 hold K+2,K+3
    const int mbase = (l >> 4) << 3;   // C: lanes 16-31 hold M=8..15

    for (int ct = 0; ct < 2; ++ct) {
        const int col = w * 32 + ct * 16 + (l & 15);   // B/D: N = lane&15
        v8f c = {};
        #pragma unroll 4
        for (int k0 = 0; k0 < DIM; k0 += 4) {
            int k = k0 + kb;
            v2f a, b;
            a.x = Ha[arow][k];           a.y = Ha[arow][k + 1];
            b.x = W1[k * DIM + col];     b.y = W1[(k + 1) * DIM + col];
            c = __builtin_amdgcn_wmma_f32_16x16x4_f32(false, a, false, b,
                                                      (short)0, c, false, false);
        }
        float bb = b1[col];
        #pragma unroll
        for (int r = 0; r < 8; ++r)
            Xs[mbase + r][col] = c[r] + bb;
    }
    __syncthreads();

    // LayerNorm partials over each 16-element segment
    {
        int row = t >> 4, seg = t & 15;
        float s = 0.f, q = 0.f;
        #pragma unroll
        for (int e = 0; e < 16; ++e) {
            float v = Xs[row][seg * 16 + e];
            s += v; q += v * v;
        }
        ps[row][seg] = s; pq[row][seg] = q;
    }
    __syncthreads();
    if (t < 16) {
        float s = 0.f, q = 0.f;
        #pragma unroll
        for (int e = 0; e < 16; ++e) { s += ps[t][e]; q += pq[t][e]; }
        float mu  = s * (1.0f / DIM);
        float var = q * (1.0f / DIM) - mu * mu;
        mu_s[t] = mu;
        sc_s[t] = rsqrtf(var + 1e-5f);
    }
    __syncthreads();
    {
        int row = t >> 4, c0 = (t & 15) * 16;
        float mu = mu_s[row], sc = sc_s[row];
        #pragma unroll
        for (int e = 0; e < 16; ++e) {
            int cidx = c0 + e;
            float v  = (Xs[row][cidx] - mu) * sc * g1[cidx] + be1[cidx];
            X[(i0 + row) * DIM + cidx] = fmaxf(v, 0.0f);
        }
    }
}

// ---------------------------------------------------------------------------
// Kernel 2: H = X @ W2 + b2   [768,32]
// ---------------------------------------------------------------------------
__global__ void __launch_bounds__(256)
k_proj2(const float* __restrict__ X, const float* __restrict__ W2,
        const float* __restrict__ b2, float* __restrict__ H)
{
    int gid = blockIdx.x * 256 + threadIdx.x;
    int n = gid >> 5, p = gid & 31;
    float acc = b2[p];
    #pragma unroll 8
    for (int k = 0; k < DIM; ++k)
        acc = fmaf(X[n * DIM + k], W2[k * PDIM + p], acc);
    H[n * PDIM + p] = acc;
}

// ---------------------------------------------------------------------------
// Kernel 3: Fi = H @ Wi + bi ; Fj = H @ Wj + bj   (blockIdx.y selects matrix)
// ---------------------------------------------------------------------------
__global__ void __launch_bounds__(256)
k_feat(const float* __restrict__ H,
       const float* __restrict__ Wi, const float* __restrict__ bi,
       const float* __restrict__ Wj, const float* __restrict__ bj,
       float* __restrict__ Fi, float* __restrict__ Fj)
{
    const float* W  = blockIdx.y ? Wj : Wi;
    const float* bv = blockIdx.y ? bj : bi;
    float*       F  = blockIdx.y ? Fj : Fi;
    int gid = blockIdx.x * 256 + threadIdx.x;
    int n = gid >> 8, c = gid & 255;
    float acc = bv[c];
    #pragma unroll
    for (int k = 0; k < PDIM; ++k)
        acc = fmaf(H[n * PDIM + k], W[k * DIM + c], acc);
    F[n * DIM + c] = acc;
}

// ---------------------------------------------------------------------------
// Kernel 4: pair stage. Per 16x16 pair tile:
//   S = Fi_tile @ Fj_tile^T via WMMA (K split over 8 waves),
//   LN stats from row sums + S, then fused mish-weighted dot per pair.
// ---------------------------------------------------------------------------
__global__ void __launch_bounds__(256)
k_pair(const float* __restrict__ Fi, const float* __restrict__ Fj,
       const float* __restrict__ g2, const float* __restrict__ be2,
       const float* __restrict__ wt, const float* __restrict__ btp,
       float* __restrict__ out)
{
    __shared__ float Fa[16][LDP];
    __shared__ float Fb[16][LDP];
    __shared__ float Sp[8][16][16];
    __shared__ float psA[16][16], pqA[16][16], psB[16][16], pqB[16][16];
    __shared__ __align__(16) float gs[256], bes[256], wts[256];

    const int t  = threadIdx.x;
    const int i0 = blockIdx.x * 16;
    const int j0 = blockIdx.y * 16;

    for (int c = 0; c < 4; ++c) {
        int chunk = t + c * 256;
        int row   = chunk >> 6;
        int col4  = (chunk & 63) << 2;
        *(float4*)(&Fa[row][col4]) = *(const float4*)(Fi + (i0 + row) * DIM + col4);
        *(float4*)(&Fb[row][col4]) = *(const float4*)(Fj + (j0 + row) * DIM + col4);
    }
    gs[t] = g2[t]; bes[t] = be2[t]; wts[t] = wt[t];
    __syncthreads();

    // per-row partial sums / sumsq (16 segs per row)
    {
        int row = t >> 4, seg = t & 15;
        float sa = 0, qa = 0, sb = 0, qb = 0;
        #pragma unroll
        for (int e = 0; e < 16; ++e) {
            float va = Fa[row][seg * 16 + e];
            float vb = Fb[row][seg * 16 + e];
            sa += va; qa += va * va; sb += vb; qb += vb * vb;
        }
        psA[row][seg] = sa; pqA[row][seg] = qa;
        psB[row][seg] = sb; pqB[row][seg] = qb;
    }

    // Gram tile S = Fa @ Fb^T via WMMA f32 16x16x4; wave w owns K=[32w,32w+32)
    {
        const int l     = t & 31;
        const int w     = t >> 5;
        const int row   = l & 15;
        const int kb    = (l >> 4) << 1;
        const int mbase = (l >> 4) << 3;
        v8f c = {};
        #pragma unroll
        for (int kk = 0; kk < 8; ++kk) {
            int k = w * 32 + kk * 4 + kb;
            v2f a, b;
            a.x = Fa[row][k]; a.y = Fa[row][k + 1];
            b.x = Fb[row][k]; b.y = Fb[row][k + 1];
            c = __builtin_amdgcn_wmma_f32_16x16x4_f32(false, a, false, b,
                                                      (short)0, c, false, false);
        }
        #pragma unroll
        for (int r = 0; r < 8; ++r)
            Sp[w][mbase + r][l & 15] = c[r];
    }
    __syncthreads();

    const int m = t >> 4, n = t & 15;
    float S = 0, sa = 0, qa = 0, sb = 0, qb = 0;
    #pragma unroll
    for (int e = 0; e < 8; ++e) S += Sp[e][m][n];
    #pragma unroll
    for (int e = 0; e < 16; ++e) {
        sa += psA[m][e]; qa += pqA[m][e];
        sb += psB[n][e]; qb += pqB[n][e];
    }
    const float inv = 1.0f / DIM;
    float mu  = (sa + sb) * inv;
    float e2  = (qa + qb + 2.0f * S) * inv;
    float var = e2 - mu * mu;
    float sc  = rsqrtf(var + 1e-5f);

    float acc = 0.0f;
    for (int k4 = 0; k4 < 64; ++k4) {
        float4 av = *(const float4*)(&Fa[m][k4 * 4]);
        float4 bv = *(const float4*)(&Fb[n][k4 * 4]);
        float4 gv = *(const float4*)(&gs[k4 * 4]);
        float4 ev = *(const float4*)(&bes[k4 * 4]);
        float4 wv = *(const float4*)(&wts[k4 * 4]);
        acc += wv.x * mish_f((av.x + bv.x - mu) * sc * gv.x + ev.x);
        acc += wv.y * mish_f((av.y + bv.y - mu) * sc * gv.y + ev.y);
        acc += wv.z * mish_f((av.z + bv.z - mu) * sc * gv.z + ev.z);
        acc += wv.w * mish_f((av.w + bv.w - mu) * sc * gv.w + ev.w);
    }
    out[(i0 + m) * NROW + (j0 + n)] = acc + btp[0];
}

// ---------------------------------------------------------------------------
extern "C" void kernel_launch(void* const* d_in, const int* in_sizes, int n_in,
                              void* d_out, int out_size, void* d_ws, size_t ws_size,
                              hipStream_t stream)
{
    const float* h   = (const float*)d_in[0];
    const float* W1  = (const float*)d_in[1];
    const float* b1  = (const float*)d_in[2];
    const float* g1  = (const float*)d_in[3];
    const float* be1 = (const float*)d_in[4];
    const float* W2  = (const float*)d_in[5];
    const float* b2  = (const float*)d_in[6];
    const float* Wi  = (const float*)d_in[7];
    const float* bi  = (const float*)d_in[8];
    const float* Wj  = (const float*)d_in[9];
    const float* bj  = (const float*)d_in[10];
    const float* g2  = (const float*)d_in[11];
    const float* be2 = (const float*)d_in[12];
    const float* Wt  = (const float*)d_in[13];
    const float* bt  = (const float*)d_in[14];

    float* X  = (float*)d_ws;              // 768*256
    float* H  = X  + NROW * DIM;           // 768*32
    float* Fi = H  + NROW * PDIM;          // 768*256
    float* Fj = Fi + NROW * DIM;           // 768*256

    k_preproj<<<NROW / 16, 256, 0, stream>>>(h, W1, b1, g1, be1, X);
    k_proj2<<<(NROW * PDIM) / 256, 256, 0, stream>>>(X, W2, b2, H);
    k_feat<<<dim3((NROW * DIM) / 256, 2), 256, 0, stream>>>(H, Wi, bi, Wj, bj, Fi, Fj);
    k_pair<<<dim3(NROW / 16, NROW / 16), 256, 0, stream>>>(Fi, Fj, g2, be2, Wt, bt,
                                                           (float*)d_out);
}